// CRFCell_910533066774
// MI455X (gfx1250) — compile-verified
//
#include <hip/hip_runtime.h>
#include <stdint.h>
#include <stddef.h>

// Viterbi CRF decode: B=512 sequences, T=512 steps, K=64 tags.
// One block (256 threads = 8 wave32) per batch element; the previous-tag
// dimension i is split across G=4 groups of 16 to shorten the serial
// per-step critical path (T-serial recurrence is the bottleneck).
// CDNA5 features: 320KB WGP LDS (transition matrix + full T*K argmax history
// resident in LDS -> no HBM round-trip for backtrace), async global->LDS
// emission prefetch (global_load_async_to_lds_b32 / s_wait_asynccnt,
// ASYNCcnt-tracked, double-buffered to overlap with the max-plus step).

#define B_  512
#define T_  512
#define K_  64
#define G_  4              // groups along previous-tag dimension
#define IPG (K_ / G_)      // 16 candidates per thread per step
#define TPB (K_ * G_)      // 256 threads = 8 wave32

// Async copy of one f32 per lane from global memory into LDS.
// VDST holds the LDS byte address (low 32 bits of the flat shared address are
// the LDS offset per the CDNA5 aperture rules), VADDR is the 64-bit global
// address. Tracked with ASYNCcnt; in-order completion per wave.
__device__ __forceinline__ void async_ld_f32_to_lds(void* lds, const float* gaddr) {
    unsigned loff = (unsigned)(size_t)lds;
    asm volatile("global_load_async_to_lds_b32 %0, %1, off"
                 :: "v"(loff), "v"(gaddr)
                 : "memory");
}

#define WAIT_ASYNC_LE_1() asm volatile("s_wait_asynccnt 1" ::: "memory")
#define WAIT_ASYNC_LE_0() asm volatile("s_wait_asynccnt 0" ::: "memory")

__global__ __launch_bounds__(TPB) void viterbi_crf_kernel(
    const float* __restrict__ em,      // [B,T,K] emissions
    const int*   __restrict__ attn,    // [B,T]   attention mask
    const float* __restrict__ startT,  // [K]     start transitions
    const float* __restrict__ endT,    // [K]     end transitions
    const float* __restrict__ trans,   // [K,K]   transitions
    int*         __restrict__ out)     // [B,T]   best tag paths (int32)
{
    __shared__ float          s_trans[K_ * K_];   // 16 KB transition matrix
    __shared__ unsigned char  s_hist[T_ * K_];    // 32 KB argmax history (row 0 unused)
    __shared__ float          s_score[K_];        // running Viterbi scores
    __shared__ float          s_ebuf[2][K_];      // double-buffered emission row
    __shared__ int            s_mask[T_];         // this sequence's mask row
    __shared__ int            s_tags[T_];         // decoded tags (staging)
    __shared__ float          s_pb[G_][K_];       // per-group partial best
    __shared__ int            s_pi[G_][K_];       // per-group partial argmax
    __shared__ float          s_red[K_];          // final-score reduction

    const int tid = threadIdx.x;
    const int j   = tid & (K_ - 1);   // next-tag index 0..63
    const int g   = tid >> 6;         // i-group 0..3 (wave-uniform)
    const int b   = blockIdx.x;       // batch element

    // ---- cooperative loads: transitions + mask row
    #pragma unroll
    for (int k = tid; k < K_ * K_; k += TPB) s_trans[k] = trans[k];
    #pragma unroll
    for (int t = tid; t < T_; t += TPB)      s_mask[t] = attn[(size_t)b * T_ + t];

    const float* em_b = em + (size_t)b * T_ * K_;

    if (tid < K_) {  // waves 0-1 (g==0): init score, prefetch emission row t=1
        s_score[j] = startT[j] + em_b[j];                    // t = 0
        async_ld_f32_to_lds(&s_ebuf[1][j], em_b + K_ + j);
    }
    __syncthreads();

    // ---- forward max-plus scan over T
    for (int t = 1; t < T_; ++t) {
        const int cur = t & 1;
        if (t + 1 < T_) {
            if (tid < K_)   // overlap: prefetch row t+1 while computing row t
                async_ld_f32_to_lds(&s_ebuf[cur ^ 1][j],
                                    em_b + (size_t)(t + 1) * K_ + j);
            WAIT_ASYNC_LE_1();   // in-order -> row t complete (issuing waves)
        } else {
            WAIT_ASYNC_LE_0();
        }
        __syncthreads();   // barrier A: emission row t + score(t-1) visible

        // partial max over 16 previous tags i in [g*16, g*16+16)
        // s_score[i]: wave-uniform LDS broadcast; s_trans[i*K+j]: stride-1.
        const int i0 = g * IPG;
        float best  = s_score[i0] + s_trans[i0 * K_ + j];
        int   besti = i0;
        #pragma unroll
        for (int r = 1; r < IPG; ++r) {
            const int i = i0 + r;
            float c = s_score[i] + s_trans[i * K_ + j];
            if (c > best) { best = c; besti = i; }   // strict > keeps first max
        }
        s_pb[g][j] = best;
        s_pi[g][j] = besti;
        __syncthreads();   // barrier B: partials visible, s_score reads done

        if (g == 0) {      // combine 4 partials; group order = ascending i
            float bb = s_pb[0][j];
            int   bi = s_pi[0][j];
            #pragma unroll
            for (int q = 1; q < G_; ++q) {
                float c = s_pb[q][j];
                if (c > bb) { bb = c; bi = s_pi[q][j]; }  // ties keep lower i
            }
            s_hist[t * K_ + j] = (unsigned char)bi;
            float ns = bb + s_ebuf[cur][j];
            ns = s_mask[t] ? ns : s_score[j];   // freeze past sequence end
            s_score[j] = ns;    // readers resync at next iteration's barrier A
        }
    }
    __syncthreads();   // final s_score writes visible

    // ---- final score + end transitions, argmax (first maximum)
    if (tid < K_) s_red[j] = s_score[j] + endT[j];
    __syncthreads();

    if (tid == 0) {
        float bv = s_red[0];
        int   bt = 0;
        for (int i = 1; i < K_; ++i)
            if (s_red[i] > bv) { bv = s_red[i]; bt = i; }

        // backtrace entirely from LDS history
        int tag = bt;
        s_tags[T_ - 1] = tag;
        for (int t = T_ - 1; t >= 1; --t) {
            int prev = (int)s_hist[t * K_ + tag];
            if (s_mask[t]) tag = prev;
            s_tags[t - 1] = tag;
        }
    }
    __syncthreads();

    // coalesced store of the decoded path
    int* outb = out + (size_t)b * T_;
    #pragma unroll
    for (int t = tid; t < T_; t += TPB) outb[t] = s_tags[t];
}

extern "C" void kernel_launch(void* const* d_in, const int* in_sizes, int n_in,
                              void* d_out, int out_size, void* d_ws, size_t ws_size,
                              hipStream_t stream) {
    (void)in_sizes; (void)n_in; (void)d_ws; (void)ws_size; (void)out_size;
    const float* em     = (const float*)d_in[0];  // [B,T,K]
    const int*   attn   = (const int*)  d_in[1];  // [B,T]
    const float* startT = (const float*)d_in[2];  // [K]
    const float* endT   = (const float*)d_in[3];  // [K]
    const float* trans  = (const float*)d_in[4];  // [K,K]
    int*         out    = (int*)d_out;            // [B,T] int32 tags

    viterbi_crf_kernel<<<B_, TPB, 0, stream>>>(em, attn, startT, endT, trans, out);
}